// ModelNew_77421080477847
// MI455X (gfx1250) — compile-verified
//
#include <hip/hip_runtime.h>
#include <stdint.h>

#define SK_ITERS 20
#define SK_EPS   1e-6f
#define SK_N     64
#define ROWW     68            // padded LDS row stride in words (272B, 16B aligned)
#define ROWB     (ROWW * 4)

// ---- CDNA5 async global <-> LDS DMA (ASYNCcnt-tracked) ----
__device__ __forceinline__ void async_load_b128(uint32_t lds_off, const void* gaddr) {
  asm volatile("global_load_async_to_lds_b128 %0, %1, off"
               :: "v"(lds_off), "v"(gaddr) : "memory");
}
__device__ __forceinline__ void async_store_b128(void* gaddr, uint32_t lds_off) {
  asm volatile("global_store_async_from_lds_b128 %0, %1, off"
               :: "v"(gaddr), "v"(lds_off) : "memory");
}
__device__ __forceinline__ void wait_async0() {
  asm volatile("s_wait_asynccnt 0" ::: "memory");
}

// One 64-thread workgroup per 64x64 matrix.
// Thread t keeps row t AND column t of X=exp(x-rowmax) in VGPRs.
// Sinkhorn state: x = diag(a) * X * diag(b); only a,b (64 floats each)
// go through LDS per iteration.
__global__ __launch_bounds__(64) void sinkhorn64_kernel(const float* __restrict__ in,
                                                        float* __restrict__ out) {
  __shared__ alignas(16) float mat[SK_N * ROWW];
  __shared__ alignas(16) float avec[SK_N];
  __shared__ alignas(16) float bvec[SK_N];

  const int t = threadIdx.x;
  const float* gin  = in  + (size_t)blockIdx.x * (SK_N * SK_N);
  float*       gout = out + (size_t)blockIdx.x * (SK_N * SK_N);

  // Low 32 bits of a generic pointer to __shared__ == LDS byte offset.
  const uint32_t ldsMat  = (uint32_t)(uintptr_t)(&mat[0]);
  // chunk k = i*64 + t -> global byte k*16; row = i*4 + (t>>4); col16 = t&15
  const uint32_t ldsLane = ldsMat + (uint32_t)((t >> 4) * ROWB + (t & 15) * 16);

  // ---- Phase 0: async DMA global -> LDS, 16 x (64 lanes x 16B) = 16KB ----
  #pragma unroll
  for (int i = 0; i < 16; ++i) {
    async_load_b128(ldsLane + (uint32_t)(i * 4 * ROWB),
                    (const char*)gin + (size_t)(i * 1024 + t * 16));
  }
  wait_async0();
  __syncthreads();

  // ---- Phase 1: row t -> regs, rowmax, exp, write back; init b ----
  float rowv[SK_N];
  #pragma unroll
  for (int i = 0; i < 16; ++i) {
    float4 v = *(const float4*)&mat[t * ROWW + i * 4];
    rowv[i*4+0] = v.x; rowv[i*4+1] = v.y; rowv[i*4+2] = v.z; rowv[i*4+3] = v.w;
  }
  float m = rowv[0];
  #pragma unroll
  for (int c = 1; c < SK_N; ++c) m = fmaxf(m, rowv[c]);
  #pragma unroll
  for (int c = 0; c < SK_N; ++c) rowv[c] = __expf(rowv[c] - m);
  #pragma unroll
  for (int i = 0; i < 16; ++i) {
    float4 v = make_float4(rowv[i*4+0], rowv[i*4+1], rowv[i*4+2], rowv[i*4+3]);
    *(float4*)&mat[t * ROWW + i * 4] = v;
  }
  bvec[t] = 1.0f;
  __syncthreads();

  // column t of X (stride-68 word layout -> conflict-free across lanes)
  float colv[SK_N];
  #pragma unroll
  for (int r = 0; r < SK_N; ++r) colv[r] = mat[r * ROWW + t];

  // ---- Phase 2: 20 Sinkhorn iterations on (a, b) only ----
  float a_t = 1.0f, b_t = 1.0f;
  for (int it = 0; it < SK_ITERS; ++it) {
    // u[t] = dot(row_t, b)   (b broadcast from LDS, 2 FMA chains for dual-issue)
    float u0 = 0.f, u1 = 0.f;
    #pragma unroll
    for (int i = 0; i < 16; ++i) {
      float4 bb = *(const float4*)&bvec[i * 4];
      u0 = fmaf(rowv[i*4+0], bb.x, u0);
      u1 = fmaf(rowv[i*4+1], bb.y, u1);
      u0 = fmaf(rowv[i*4+2], bb.z, u0);
      u1 = fmaf(rowv[i*4+3], bb.w, u1);
    }
    float u = u0 + u1;
    // a = a / (a*u + eps)  via v_rcp_f32
    a_t = a_t * __builtin_amdgcn_rcpf(fmaf(a_t, u, SK_EPS));
    avec[t] = a_t;
    __syncthreads();

    // w[t] = dot(col_t, a')
    float w0 = 0.f, w1 = 0.f;
    #pragma unroll
    for (int i = 0; i < 16; ++i) {
      float4 aa = *(const float4*)&avec[i * 4];
      w0 = fmaf(colv[i*4+0], aa.x, w0);
      w1 = fmaf(colv[i*4+1], aa.y, w1);
      w0 = fmaf(colv[i*4+2], aa.z, w0);
      w1 = fmaf(colv[i*4+3], aa.w, w1);
    }
    float w = w0 + w1;
    b_t = b_t * __builtin_amdgcn_rcpf(fmaf(b_t, w, SK_EPS));
    bvec[t] = b_t;
    __syncthreads();
  }

  // ---- Phase 3: out[r][c] = X[r][c]*a[r]*b[c]; stage in LDS, async DMA out ----
  #pragma unroll
  for (int i = 0; i < 16; ++i) {
    float4 bb = *(const float4*)&bvec[i * 4];
    float4 o;
    o.x = rowv[i*4+0] * a_t * bb.x;
    o.y = rowv[i*4+1] * a_t * bb.y;
    o.z = rowv[i*4+2] * a_t * bb.z;
    o.w = rowv[i*4+3] * a_t * bb.w;
    *(float4*)&mat[t * ROWW + i * 4] = o;
  }
  __syncthreads();
  #pragma unroll
  for (int i = 0; i < 16; ++i) {
    async_store_b128((char*)gout + (size_t)(i * 1024 + t * 16),
                     ldsLane + (uint32_t)(i * 4 * ROWB));
  }
  wait_async0();
}

extern "C" void kernel_launch(void* const* d_in, const int* in_sizes, int n_in,
                              void* d_out, int out_size, void* d_ws, size_t ws_size,
                              hipStream_t stream) {
  (void)n_in; (void)out_size; (void)d_ws; (void)ws_size;
  const float* logits = (const float*)d_in[0];
  float* out = (float*)d_out;
  const int nmat = in_sizes[0] / (SK_N * SK_N);   // 128*64 = 8192 matrices
  sinkhorn64_kernel<<<dim3(nmat), dim3(SK_N), 0, stream>>>(logits, out);
}